// SNE_45526653338152
// MI455X (gfx1250) — compile-verified
//
#include <hip/hip_runtime.h>
#include <stdint.h>

// SNE (surface normal estimation) stencil for MI455X / gfx1250.
// Strategy: async-copy depth tile(+halo) global->LDS (CDNA5 ASYNCcnt path),
// compute X/Y/Z/D once per cell into LDS, then per-pixel 8-direction stencil.
// Outputs streamed with non-temporal stores (write-once, 25 MB).

#define IMG_H 1080
#define IMG_W 1920
#define BW 32
#define BH 8
#define SW (BW + 2)          // 34
#define SH (BH + 2)          // 10
#define SN (SW * SH)         // 340
#define SN_PAD 512           // padded so halo staging runs 2 full-EXEC rounds
#define NTHREADS (BW * BH)   // 256 threads = 8 wave32
#define PI_F 3.141592657f    // exact constant from the reference

__global__ __launch_bounds__(NTHREADS)
void SNE_45526653338152_kernel(const float* __restrict__ depth,
                               const float* __restrict__ cam,
                               float* __restrict__ out)
{
    __shared__ float sRaw[SN_PAD];   // raw depth tile + halo (async target)
    __shared__ float sD[SN];         // 1/Z
    __shared__ float sX[SN];
    __shared__ float sY[SN];
    __shared__ float sZ[SN];

    const int tx  = threadIdx.x;            // 0..31
    const int ty  = threadIdx.y;            // 0..7
    const int tid = ty * BW + tx;           // 0..255
    const int bx0 = blockIdx.x * BW;
    const int by0 = blockIdx.y * BH;

    const float fx = cam[0];
    const float fy = cam[4];
    const float cx = cam[2];
    const float cy = cam[5];
    const float inv_fx = __builtin_amdgcn_rcpf(fx);

    // ---- Phase 1: async global->LDS staging of the depth halo tile ----
    // Addresses clamped in-bounds so every lane issues (full EXEC); cells
    // outside the image are overridden with zeros in phase 2.
    #pragma unroll
    for (int idx = tid; idx < SN_PAD; idx += NTHREADS) {
        const int r  = idx / SW;
        const int c  = idx - r * SW;
        int gy = by0 + r - 1;
        int gx = bx0 + c - 1;
        gy = gy < 0 ? 0 : (gy >= IMG_H ? IMG_H - 1 : gy);
        gx = gx < 0 ? 0 : (gx >= IMG_W ? IMG_W - 1 : gx);
        const float* gptr = depth + (size_t)gy * IMG_W + gx;
        const uint32_t loff = (uint32_t)(size_t)(const void*)&sRaw[idx];
        // CDNA5 async copy: copies 4 bytes from the global address in gptr to the LDS address in loff; tracked by ASYNCcnt.
        asm volatile("global_load_async_to_lds_b32 %0, %1, off"
                     :: "v"(loff), "v"(gptr)
                     : "memory");
    }
#if __has_builtin(__builtin_amdgcn_s_wait_asynccnt)
    __builtin_amdgcn_s_wait_asynccnt(0);
#else
    asm volatile("s_wait_asynccnt 0" ::: "memory");
#endif
    __syncthreads();

    // ---- Phase 2: compute X, Y, Z, D = 1/Z once per cell (shared) ----
    for (int idx = tid; idx < SN; idx += NTHREADS) {
        const int r  = idx / SW;
        const int c  = idx - r * SW;
        const int gy = by0 + r - 1;
        const int gx = bx0 + c - 1;
        float Xv = 0.0f, Yv = 0.0f, Zv = 0.0f, Dv = 0.0f;
        if ((unsigned)gy < (unsigned)IMG_H && (unsigned)gx < (unsigned)IMG_W) {
            const float Z0 = sRaw[idx];
            float Yt = Z0 * ((float)gy - cy) * inv_fx;   // ref divides by fx for both
            float Xt = Z0 * ((float)gx - cx) * inv_fx;
            const bool neg = (Yt <= 0.0f);
            float Zt = neg ? 0.0f : Z0;
            Yt       = neg ? 0.0f : Yt;
            Zt = (Zt == Zt) ? Zt : 0.0f;                 // NaN -> 0
            Xv = Xt; Yv = Yt; Zv = Zt;
            Dv = __builtin_amdgcn_rcpf(Zt);              // 1/0 -> +inf (matches torch)
        }
        sX[idx] = Xv; sY[idx] = Yv; sZ[idx] = Zv; sD[idx] = Dv;
    }
    __syncthreads();

    // ---- Phase 3: per-pixel stencil ----
    const int gx = bx0 + tx;
    const int gy = by0 + ty;
    const int sc = (ty + 1) * SW + (tx + 1);

    const float nx_t = (sD[sc + 1]  - sD[sc - 1])  * fx;  // Gx * fx
    const float ny_t = (sD[sc + SW] - sD[sc - SW]) * fy;  // Gy * fy
    const float phi  = atanf(ny_t * __builtin_amdgcn_rcpf(nx_t)) + PI_F;
    const float a    = __cosf(phi);
    const float b    = __sinf(phi);

    const float Xc = sX[sc], Yc = sY[sc], Zc = sZ[sc];
    const float c2 = nx_t * nx_t + ny_t * ny_t;          // hoisted from loop

    float snx = 0.0f, sny = 0.0f, snz = 0.0f;
    const int offs[8] = {-SW - 1, -SW, -SW + 1, -1, +1, SW - 1, SW, SW + 1};
    #pragma unroll
    for (int k = 0; k < 8; ++k) {
        const int o = sc + offs[k];
        const float Xd = Xc - sX[o];
        const float Yd = Yc - sY[o];
        const float Zd = Zc - sZ[o];
        const float nz_i = (nx_t * Xd + ny_t * Yd) * __builtin_amdgcn_rcpf(Zd);
        const float inv  = __builtin_amdgcn_rsqf(c2 + nz_i * nz_i); // 1/norm
        const float nxi = nx_t * inv;
        const float nyi = ny_t * inv;
        const float nzi = nz_i * inv;
        snx += (nxi == nxi) ? nxi : 0.0f;    // NaN -> 0 before sum
        sny += (nyi == nyi) ? nyi : 0.0f;
        snz += (nzi == nzi) ? nzi : 0.0f;
    }

    const float theta = -atanf((snx * a + sny * b) * __builtin_amdgcn_rcpf(snz));
    const float st = __sinf(theta);
    float nxo = st * a;                       // sin(theta)*cos(phi)
    float nyo = st * b;                       // sin(theta)*sin(phi)
    float nzo = __cosf(theta);
    if (!(nzo == nzo)) { nxo = 0.0f; nyo = 0.0f; nzo = -1.0f; }
    const float sgn = (nyo > 0.0f) ? -1.0f : 1.0f;

    if (gx < IMG_W && gy < IMG_H) {
        const size_t hw = (size_t)IMG_H * IMG_W;
        const size_t p  = (size_t)gy * IMG_W + gx;
        __builtin_nontemporal_store(nxo * sgn, out + p);
        __builtin_nontemporal_store(nyo * sgn, out + hw + p);
        __builtin_nontemporal_store(nzo * sgn, out + 2 * hw + p);
    }
}

extern "C" void kernel_launch(void* const* d_in, const int* in_sizes, int n_in,
                              void* d_out, int out_size, void* d_ws, size_t ws_size,
                              hipStream_t stream) {
    const float* depth = (const float*)d_in[0];   // [1080,1920] f32
    const float* cam   = (const float*)d_in[1];   // [3,3] f32
    float* out = (float*)d_out;                   // [3,1080,1920] f32
    (void)in_sizes; (void)n_in; (void)out_size; (void)d_ws; (void)ws_size;

    dim3 block(BW, BH);
    dim3 grid((IMG_W + BW - 1) / BW, (IMG_H + BH - 1) / BH);  // 60 x 135
    hipLaunchKernelGGL(SNE_45526653338152_kernel, grid, block, 0, stream,
                       depth, cam, out);
}